// SparseGraphAttnLayer_16561393893396
// MI455X (gfx1250) — compile-verified
//
#include <hip/hip_runtime.h>
#include <hip/hip_bf16.h>

#define GAT_ALPHA 0.2f
#define ORD_NEG_INF 0x007FFFFFu   // order-encoded -inf

typedef __attribute__((ext_vector_type(2))) float v2f;
typedef __attribute__((ext_vector_type(8))) float v8f;

// Monotonic float <-> uint mapping so atomicMax(uint) implements float max.
__device__ __forceinline__ unsigned f2ord(float x) {
    unsigned u = __float_as_uint(x);
    return (u & 0x80000000u) ? ~u : (u | 0x80000000u);
}
__device__ __forceinline__ float ord2f(unsigned o) {
    unsigned u = (o & 0x80000000u) ? (o & 0x7FFFFFFFu) : ~o;
    return __uint_as_float(u);
}
__device__ __forceinline__ float lrelu(float e) {
    return e > 0.0f ? e : GAT_ALPHA * e;
}

// ---------------------------------------------------------------------------
// Kernel 0: initialize out (zeros), s (zeros), m_ord (-inf encoded)
// ---------------------------------------------------------------------------
__global__ void gat_init(float* __restrict__ out, float* __restrict__ s,
                         unsigned* __restrict__ m_ord, int N) {
    int tid = blockIdx.x * blockDim.x + threadIdx.x;
    long total = (long)N * 64;
    if (tid < total) out[tid] = 0.0f;
    if (tid < N) { s[tid] = 0.0f; m_ord[tid] = ORD_NEG_INF; }
}

// ---------------------------------------------------------------------------
// Kernel 1: h = x @ W via V_WMMA_F32_16X16X4_F32.
// One wave per 16-row strip; 4 accumulators cover the 64 output columns.
// A 16x4 layout : lane<16 -> K = k+0,k+1 ; lane>=16 -> K = k+2,k+3 ; M = lane&15
// B  4x16 layout: same K split; N = lane&15
// C/D 16x16     : VGPR v -> row (lane<16 ? v : v+8), col = lane&15
// ---------------------------------------------------------------------------
__global__ __launch_bounds__(32) void gat_gemm_h(const float* __restrict__ x,
                                                 const float* __restrict__ W,
                                                 float* __restrict__ h, int N) {
    const int lane   = threadIdx.x;          // 0..31 (wave32)
    const int mLocal = lane & 15;
    const int kOff   = (lane < 16) ? 0 : 2;
    int row = blockIdx.x * 16 + mLocal;
    int rowClamped = row < N ? row : (N - 1); // keep loads in-bounds, EXEC uniform
    const float* xrow = x + (size_t)rowClamped * 128;

    v8f c0 = {}, c1 = {}, c2 = {}, c3 = {};
    #pragma unroll 4
    for (int k = 0; k < 128; k += 4) {
        v2f a;
        a.x = xrow[k + kOff + 0];
        a.y = xrow[k + kOff + 1];
        const float* w0 = W + (size_t)(k + kOff + 0) * 64;
        const float* w1 = W + (size_t)(k + kOff + 1) * 64;
        v2f b0, b1, b2, b3;
        b0.x = w0[mLocal +  0]; b0.y = w1[mLocal +  0];
        b1.x = w0[mLocal + 16]; b1.y = w1[mLocal + 16];
        b2.x = w0[mLocal + 32]; b2.y = w1[mLocal + 32];
        b3.x = w0[mLocal + 48]; b3.y = w1[mLocal + 48];
        c0 = __builtin_amdgcn_wmma_f32_16x16x4_f32(false, a, false, b0, (short)0, c0, false, false);
        c1 = __builtin_amdgcn_wmma_f32_16x16x4_f32(false, a, false, b1, (short)0, c1, false, false);
        c2 = __builtin_amdgcn_wmma_f32_16x16x4_f32(false, a, false, b2, (short)0, c2, false, false);
        c3 = __builtin_amdgcn_wmma_f32_16x16x4_f32(false, a, false, b3, (short)0, c3, false, false);
    }

    const int rBase = blockIdx.x * 16 + ((lane < 16) ? 0 : 8);
    #pragma unroll
    for (int v = 0; v < 8; ++v) {
        int r = rBase + v;
        if (r < N) {
            float* hr = h + (size_t)r * 64 + mLocal;
            hr[ 0] = c0[v];
            hr[16] = c1[v];
            hr[32] = c2[v];
            hr[48] = c3[v];
        }
    }
}

// ---------------------------------------------------------------------------
// Kernel 2: ar[i] = h[i,:]·a_row, ac[i] = h[i,:]·a_col  (one wave per node)
// ---------------------------------------------------------------------------
__global__ void gat_rowscores(const float* __restrict__ h,
                              const float* __restrict__ a_row,
                              const float* __restrict__ a_col,
                              float* __restrict__ ar, float* __restrict__ ac, int N) {
    int node = blockIdx.x * (blockDim.x / 32) + (threadIdx.x / 32);
    int lane = threadIdx.x & 31;
    if (node >= N) return;
    const float2 hv = *(const float2*)(h + (size_t)node * 64 + lane * 2);
    float pr = hv.x * a_row[lane * 2] + hv.y * a_row[lane * 2 + 1];
    float pc = hv.x * a_col[lane * 2] + hv.y * a_col[lane * 2 + 1];
    #pragma unroll
    for (int off = 16; off > 0; off >>= 1) {
        pr += __shfl_down(pr, off, 32);
        pc += __shfl_down(pc, off, 32);
    }
    if (lane == 0) { ar[node] = pr; ac[node] = pc; }
}

// ---------------------------------------------------------------------------
// Kernel 3: segment max of e over destination rows (ordered-uint atomicMax)
// ---------------------------------------------------------------------------
__global__ void gat_edge_max(const int* __restrict__ rows, const int* __restrict__ cols,
                             const float* __restrict__ ar, const float* __restrict__ ac,
                             unsigned* __restrict__ m_ord, int E) {
    int e = blockIdx.x * blockDim.x + threadIdx.x;
    if (e >= E) return;
    int r = rows[e], c = cols[e];
    float ev = lrelu(ar[r] + ac[c]);
    atomicMax(&m_ord[r], f2ord(ev));
}

// ---------------------------------------------------------------------------
// Kernel 4: s[r] += exp(e - m[r])   (recompute e, no edge storage)
// ---------------------------------------------------------------------------
__global__ void gat_edge_sum(const int* __restrict__ rows, const int* __restrict__ cols,
                             const float* __restrict__ ar, const float* __restrict__ ac,
                             const unsigned* __restrict__ m_ord, float* __restrict__ s, int E) {
    int e = blockIdx.x * blockDim.x + threadIdx.x;
    if (e >= E) return;
    int r = rows[e], c = cols[e];
    float ev = lrelu(ar[r] + ac[c]);
    float ex = __expf(ev - ord2f(m_ord[r]));
    atomicAdd(&s[r], ex);
}

// ---------------------------------------------------------------------------
// Kernel 5: out[r,:] += attn_e * h[c,:]   (one wave per edge, 2 feats/lane)
// ---------------------------------------------------------------------------
__global__ void gat_spmm(const int* __restrict__ rows, const int* __restrict__ cols,
                         const float* __restrict__ ar, const float* __restrict__ ac,
                         const unsigned* __restrict__ m_ord, const float* __restrict__ s,
                         const float* __restrict__ h, float* __restrict__ out, int E) {
    int e = blockIdx.x * (blockDim.x / 32) + (threadIdx.x / 32);
    if (e >= E) return;
    int lane = threadIdx.x & 31;
    int r = rows[e], c = cols[e];
    float ev = lrelu(ar[r] + ac[c]);
    float attn = __expf(ev - ord2f(m_ord[r])) / s[r];
    const float2 hv = *(const float2*)(h + (size_t)c * 64 + lane * 2);
    float* orow = out + (size_t)r * 64 + lane * 2;
    atomicAdd(orow + 0, attn * hv.x);
    atomicAdd(orow + 1, attn * hv.y);
}

// ---------------------------------------------------------------------------
extern "C" void kernel_launch(void* const* d_in, const int* in_sizes, int n_in,
                              void* d_out, int out_size, void* d_ws, size_t ws_size,
                              hipStream_t stream) {
    const float* x     = (const float*)d_in[0];
    const int*   rows  = (const int*)  d_in[1];
    const int*   cols  = (const int*)  d_in[2];
    const float* W     = (const float*)d_in[3];
    const float* a_row = (const float*)d_in[4];
    const float* a_col = (const float*)d_in[5];

    const int D = in_sizes[4];            // 64
    const int F = in_sizes[3] / D;        // 128
    const int N = in_sizes[0] / F;        // 100000
    const int E = in_sizes[1];            // 1600000
    (void)F; (void)ws_size; (void)n_in;

    float*    ws    = (float*)d_ws;
    float*    h     = ws;                          // N*64
    float*    ar    = h  + (size_t)N * 64;         // N
    float*    ac    = ar + N;                      // N
    unsigned* m_ord = (unsigned*)(ac + N);         // N
    float*    s     = (float*)(m_ord + N);         // N
    float*    out   = (float*)d_out;
    (void)out_size;

    // 0) init out / s / m
    {
        long total = (long)N * 64;
        int blocks = (int)((total + 255) / 256);
        gat_init<<<blocks, 256, 0, stream>>>(out, s, m_ord, N);
    }
    // 1) h = x @ W  (WMMA f32)
    {
        int tiles = (N + 15) / 16;
        gat_gemm_h<<<tiles, 32, 0, stream>>>(x, W, h, N);
    }
    // 2) ar / ac
    {
        int wavesPerBlock = 8;  // 256 threads
        int blocks = (N + wavesPerBlock - 1) / wavesPerBlock;
        gat_rowscores<<<blocks, 256, 0, stream>>>(h, a_row, a_col, ar, ac, N);
    }
    // 3) segment max
    gat_edge_max<<<(E + 255) / 256, 256, 0, stream>>>(rows, cols, ar, ac, m_ord, E);
    // 4) segment sum of exp
    gat_edge_sum<<<(E + 255) / 256, 256, 0, stream>>>(rows, cols, ar, ac, m_ord, s, E);
    // 5) SpMM with attention weights
    {
        int edgesPerBlock = 8;  // 256 threads = 8 waves
        int blocks = (E + edgesPerBlock - 1) / edgesPerBlock;
        gat_spmm<<<blocks, 256, 0, stream>>>(rows, cols, ar, ac, m_ord, s, h, out, E);
    }
}